// SinkhornPermutation_22849226014935
// MI455X (gfx1250) — compile-verified
//
#include <hip/hip_runtime.h>
#include <math.h>

#define NS 16
#define NN 1024
#define MB 256
#define NITERS 21
#define LOG2E 1.4426950408889634f
#define EPSC 1e-10f

typedef float v2f __attribute__((ext_vector_type(2)));
typedef float v8f __attribute__((ext_vector_type(8)));

__device__ __forceinline__ float fexp2(float x) { return __builtin_amdgcn_exp2f(x); }
__device__ __forceinline__ float flog2(float x) { return __builtin_amdgcn_logf(x); }

// a2[s][i][j] = log2(e) * (log_alpha[i][j] - log(EPS - log(u[s][i][j] + EPS)));  r2 init 0
__global__ void __launch_bounds__(256) prep_kernel(const float* __restrict__ la,
                                                   const float* __restrict__ u,
                                                   float* __restrict__ a2,
                                                   float* __restrict__ r2) {
    unsigned idx = blockIdx.x * 256u + threadIdx.x;       // 0 .. 16M-1
    unsigned ij  = idx & (NN * NN - 1);
    float uu    = u[idx];
    float noise = -logf(EPSC - logf(uu + EPSC));
    a2[idx] = LOG2E * (la[ij] + noise);
    if (idx < NS * NN) r2[idx] = 0.0f;
}

// partial col sums: parts[rc][s][j] = sum_{i in chunk} 2^(a2[s][i][j] - r2[s][i])
// grid (4 colchunks, 8 rowchunks, 16 samples), block 256 (thread-per-column)
__global__ void __launch_bounds__(256) colpass_kernel(const float* __restrict__ a2,
                                                      const float* __restrict__ r2,
                                                      float* __restrict__ parts) {
    __shared__ float rsh[128];
    const int s = blockIdx.z, rc = blockIdx.y, cc = blockIdx.x;
    const int t = threadIdx.x;
    const int j = cc * 256 + t;
    const int i0 = rc * 128;
    if (t < 128) rsh[t] = r2[s * NN + i0 + t];
    __syncthreads();
    const float* ap = a2 + ((size_t)s * NN + i0) * NN + j;
    float s0 = 0.f, s1 = 0.f, s2 = 0.f, s3 = 0.f;
    for (int i = 0; i < 128; i += 4) {
        s0 += fexp2(ap[(size_t)(i + 0) * NN] - rsh[i + 0]);
        s1 += fexp2(ap[(size_t)(i + 1) * NN] - rsh[i + 1]);
        s2 += fexp2(ap[(size_t)(i + 2) * NN] - rsh[i + 2]);
        s3 += fexp2(ap[(size_t)(i + 3) * NN] - rsh[i + 3]);
    }
    parts[((size_t)rc * NS + s) * NN + j] = (s0 + s1) + (s2 + s3);
}

// c2[s][j] = log2( sum_p parts[p][s][j] );  grid 64 x 256
__global__ void __launch_bounds__(256) colfin_kernel(const float* __restrict__ parts,
                                                     float* __restrict__ c2) {
    const int idx = blockIdx.x * 256 + threadIdx.x;   // s*NN + j
    float sum = 0.f;
#pragma unroll
    for (int p = 0; p < 8; ++p) sum += parts[(size_t)p * NS * NN + idx];
    c2[idx] = flog2(sum);
}

// r2[s][i] = log2( sum_j 2^(a2[s][i][j] - c2[s][j]) )
// grid (128 rowblocks, 16 samples), block 256 = 8 waves, wave-per-row
__global__ void __launch_bounds__(256) rowpass_kernel(const float* __restrict__ a2,
                                                      const float* __restrict__ c2,
                                                      float* __restrict__ r2) {
    __shared__ float csh[NN];
    const int s = blockIdx.y;
    const int t = threadIdx.x;
    for (int q = t; q < NN; q += 256) csh[q] = c2[s * NN + q];
    __syncthreads();
    const int w = t >> 5, l = t & 31;
    const int i = blockIdx.x * 8 + w;
    const float* ap = a2 + ((size_t)s * NN + i) * NN;
    float s0 = 0.f, s1 = 0.f;
    for (int j = l; j < NN; j += 64) {
        s0 += fexp2(ap[j]      - csh[j]);
        s1 += fexp2(ap[j + 32] - csh[j + 32]);
    }
    float sum = s0 + s1;
#pragma unroll
    for (int m = 16; m >= 1; m >>= 1) sum += __shfl_xor(sum, m, 32);
    if (l == 0) r2[s * NN + i] = flog2(sum);
}

// out[s] = x @ P_s,  P = 2^(a2 - r2 - c2),  via V_WMMA_F32_16X16X4_F32
// grid (16 col-bands of 64, 16 samples), block 256 = 8 waves; wave w owns rows [w*32, w*32+32)
__global__ void __launch_bounds__(256) gemm_kernel(const float* __restrict__ x,
                                                   const float* __restrict__ a2,
                                                   const float* __restrict__ r2,
                                                   const float* __restrict__ c2,
                                                   float* __restrict__ out) {
    constexpr int XS = 34;               // padded LDS stride for x tile rows
    constexpr int PT = 34;               // padded LDS stride for transposed P tile (per column)
    __shared__ float xs[MB * XS];        // 256 x 32 chunk of x        (34816 B)
    __shared__ float ps[64 * PT];        // 64 cols x 32 k, TRANSPOSED ( 8704 B)
    const int s  = blockIdx.y;
    const int m0 = blockIdx.x * 64;
    const int t  = threadIdx.x;
    const int w  = t >> 5, l = t & 31;
    const int lm = l & 15, lh = l >> 4;  // lane%16, lane/16

    const float* ra = a2 + (size_t)s * NN * NN;
    const float* rr = r2 + s * NN;
    const float* rc = c2 + s * NN;

    v8f acc[2][4];
#pragma unroll
    for (int rt = 0; rt < 2; ++rt)
#pragma unroll
        for (int ct = 0; ct < 4; ++ct)
#pragma unroll
            for (int g = 0; g < 8; ++g) acc[rt][ct][g] = 0.0f;

    for (int k0 = 0; k0 < NN; k0 += 32) {
        __syncthreads();
        // stage x[0:256][k0:k0+32]  (row-major, padded stride)
#pragma unroll
        for (int it = 0; it < 32; ++it) {
            int q = it * 256 + t;
            int row = q >> 5, kk = q & 31;
            xs[row * XS + kk] = x[row * NN + k0 + kk];
        }
        // stage P[k0:k0+32][m0:m0+64] = 2^(a2 - r - c), TRANSPOSED: ps[m][kk]
        // so a B fragment (K, K+1) for a fixed column is one contiguous b64.
#pragma unroll
        for (int it = 0; it < 8; ++it) {
            int q = it * 256 + t;
            int kk = q >> 6, m = q & 63;
            ps[m * PT + kk] =
                fexp2(ra[(size_t)(k0 + kk) * NN + m0 + m] - rr[k0 + kk] - rc[m0 + m]);
        }
        __syncthreads();
#pragma unroll
        for (int ks = 0; ks < 32; ks += 4) {
            const int ka = ks + (lh << 1);        // (K, K+1) per ISA 16x4 f32 A/B layout
            v2f a[2];
#pragma unroll
            for (int rt = 0; rt < 2; ++rt) {
                int row = w * 32 + rt * 16 + lm;
                a[rt] = *(const v2f*)&xs[row * XS + ka];
            }
#pragma unroll
            for (int ct = 0; ct < 4; ++ct) {
                int col = ct * 16 + lm;
                v2f b = *(const v2f*)&ps[col * PT + ka];
#pragma unroll
                for (int rt = 0; rt < 2; ++rt) {
                    acc[rt][ct] = __builtin_amdgcn_wmma_f32_16x16x4_f32(
                        false, a[rt], false, b, (short)0, acc[rt][ct], false, false);
                }
            }
        }
    }
    float* ro = out + (size_t)s * MB * NN;
#pragma unroll
    for (int rt = 0; rt < 2; ++rt)
#pragma unroll
        for (int ct = 0; ct < 4; ++ct)
#pragma unroll
            for (int g = 0; g < 8; ++g) {
                int row = w * 32 + rt * 16 + lh * 8 + g;
                int col = m0 + ct * 16 + lm;
                ro[row * NN + col] = acc[rt][ct][g];
            }
}

extern "C" void kernel_launch(void* const* d_in, const int* in_sizes, int n_in,
                              void* d_out, int out_size, void* d_ws, size_t ws_size,
                              hipStream_t stream) {
    const float* x  = (const float*)d_in[0];   // (256, 1024)
    const float* la = (const float*)d_in[1];   // (1024, 1024)
    const float* u  = (const float*)d_in[2];   // (16, 1024, 1024)
    float* out = (float*)d_out;                // (16, 256, 1024)

    // workspace layout: a2 (64 MB) | r2 (64 KB) | c2 (64 KB) | parts (512 KB)
    char* ws = (char*)d_ws;
    float* a2    = (float*)ws;
    float* r2    = (float*)(ws + (size_t)NS * NN * NN * sizeof(float));
    float* c2    = r2 + NS * NN;
    float* parts = c2 + NS * NN;

    prep_kernel<<<(NS * NN * NN) / 256, 256, 0, stream>>>(la, u, a2, r2);
    for (int it = 0; it < NITERS; ++it) {
        colpass_kernel<<<dim3(4, 8, NS), 256, 0, stream>>>(a2, r2, parts);
        colfin_kernel<<<(NS * NN) / 256, 256, 0, stream>>>(parts, c2);
        rowpass_kernel<<<dim3(NN / 8, NS), 256, 0, stream>>>(a2, c2, r2);
    }
    gemm_kernel<<<dim3(NN / 64, NS), 256, 0, stream>>>(x, a2, r2, c2, out);
}